// TR_C620_53008486367235
// MI455X (gfx1250) — compile-verified
//
#include <hip/hip_runtime.h>

// sp2wt: w_k = 100 * (x .* s_k) / clip(rowsum(x .* s_k), 1e-8, 1e8), k=1..4
// N x 64 fp32, pure streaming: 576MB total traffic -> ~25us at 23.3 TB/s.
// Strategy: 16 lanes per row (float4/lane), wave32 ds_swizzle XOR reduction,
// non-temporal b128 loads/stores in SADDR form (scalar base + zext 32-bit
// voffset + constant IOFFSET), 8-deep unrolled block-strided loop.

typedef float v4f __attribute__((ext_vector_type(4)));

// ds_swizzle group-of-32 mode: offset = {1'b0, xor[14:10], or[9:5], and[4:0]}
// and_mask=0x1f, or=0, xor=MASK  ->  lane' = lane ^ MASK (stays within 16-lane
// halves for MASK in {1,2,4,8}).
template <int MASK>
__device__ __forceinline__ float xor_add(float v) {
  constexpr int pat = (MASK << 10) | 0x1f;
  return v + __int_as_float(__builtin_amdgcn_ds_swizzle(__float_as_int(v), pat));
}

// Full reduction across the 16-lane group that owns one row.
__device__ __forceinline__ float rowsum16(float v) {
  v = xor_add<1>(v);
  v = xor_add<2>(v);
  v = xor_add<4>(v);
  v = xor_add<8>(v);
  return v;
}

// Unsigned byte offset on a char* base -> zext(i32) GEP -> SADDR-form
// global_load/store (SGPR base pair + 32-bit VGPR offset + imm IOFFSET).
__device__ __forceinline__ v4f ld_nt(const float* p, unsigned boff) {
  return __builtin_nontemporal_load(
      (const v4f*)((const char*)p + boff));
}
__device__ __forceinline__ void st_nt(float* p, unsigned boff, v4f v) {
  __builtin_nontemporal_store(v, (v4f*)((char*)p + boff));
}

#define ITERS 8
#define BLOCK 256

__global__ __launch_bounds__(BLOCK) void sp2wt_kernel(
    const float* __restrict__ x,
    const float* __restrict__ s1, const float* __restrict__ s2,
    const float* __restrict__ s3, const float* __restrict__ s4,
    float* __restrict__ o1, float* __restrict__ o2,
    float* __restrict__ o3, float* __restrict__ o4,
    unsigned total_bytes) {
  // Each block owns a contiguous chunk of BLOCK*ITERS quads (16B each).
  const unsigned boff0 =
      (blockIdx.x * (BLOCK * ITERS) + threadIdx.x) * 16u;

#pragma unroll
  for (int i = 0; i < ITERS; ++i) {
    const unsigned boff = boff0 + (unsigned)(i * BLOCK * 16);  // imm-foldable
    if (boff >= total_bytes) break;

    // x is read once and reused for all four s streams.
    const v4f xv = ld_nt(x, boff);

    v4f sv, p, o;
    float b, sc;

#define DO_S(SP, OP)                              \
    sv = ld_nt((SP), boff);                       \
    p = xv * sv;                                  \
    b = rowsum16(p.x + p.y + p.z + p.w);          \
    b = fminf(fmaxf(b, 1e-8f), 1e8f);             \
    sc = 100.0f / b;                              \
    o = p * sc;                                   \
    st_nt((OP), boff, o);

    DO_S(s1, o1)
    DO_S(s2, o2)
    DO_S(s3, o3)
    DO_S(s4, o4)
#undef DO_S
  }
}

extern "C" void kernel_launch(void* const* d_in, const int* in_sizes, int n_in,
                              void* d_out, int out_size, void* d_ws, size_t ws_size,
                              hipStream_t stream) {
  const float* x  = (const float*)d_in[0];
  const float* s1 = (const float*)d_in[1];
  const float* s2 = (const float*)d_in[2];
  const float* s3 = (const float*)d_in[3];
  const float* s4 = (const float*)d_in[4];
  float* out = (float*)d_out;

  const int n_elems = in_sizes[0];                 // N * 64 = 16,777,216
  float* o1 = out;
  float* o2 = out + (size_t)n_elems;
  float* o3 = out + (size_t)2 * n_elems;
  float* o4 = out + (size_t)3 * n_elems;

  const unsigned total_bytes = (unsigned)n_elems * 4u;   // 64MB per array
  const int total_quads = n_elems >> 2;                  // 4,194,304
  const int quads_per_block = BLOCK * ITERS;             // 2048
  const int blocks = (total_quads + quads_per_block - 1) / quads_per_block;

  sp2wt_kernel<<<blocks, BLOCK, 0, stream>>>(x, s1, s2, s3, s4,
                                             o1, o2, o3, o4, total_bytes);
}